// FDK_48103633715545
// MI455X (gfx1250) — compile-verified
//
#include <hip/hip_runtime.h>
#include <math.h>

typedef float v2f __attribute__((ext_vector_type(2)));
typedef float v8f __attribute__((ext_vector_type(8)));

#define NAE 64
#define NZE 8
#define NYE 256
#define NXE 256
#define NUE 512
#define NVE 8
#define LDSPAD 4

// ---------------------------------------------------------------------------
// Chunk kernel: block (a, c) accumulates y in [c*ychunk, (c+1)*ychunk) of the
// per-angle GEMM  D[v,u] = sum_{y,z} W_y[v,z] * Q_y[z,u]  using
// v_wmma_f32_16x16x4_f32 (M=16 padded, N=512 in 16-wide tiles, K=4 z-slices).
// Raw partial sums (no DIST) written to partial[((c*NA + a)*NV + v)*NU + u].
// 256 threads = 8 wave32; wave w owns u-tiles w*4 .. w*4+3.
// ---------------------------------------------------------------------------
__global__ __launch_bounds__(256) void fdk_chunk_kernel(const float* __restrict__ vol,
                                                        float* __restrict__ partial,
                                                        int ychunk) {
    // double-buffered rotated (z,x) plane -> one barrier per y-iteration
    __shared__ float P[2][NZE][NXE + LDSPAD];

    const int tid  = threadIdx.x;
    const int lane = tid & 31;
    const int wave = tid >> 5;
    const int nidx = lane & 15;   // N (u) index in tile / M (v) row for A-frag
    const int half = lane >> 4;   // half-wave selects K pair {0,1} vs {2,3}
    const int a    = blockIdx.x;
    const int c    = blockIdx.y;
    const int y0   = c * ychunk;
    const int y1   = y0 + ychunk;

    // Reference: cs = -sin(theta), sn = cos(theta)
    const float theta = (float)a * 0.09817477042468103f;  // 2*pi/64
    const float sth = sinf(theta);
    const float cth = cosf(theta);
    const float cs = -sth;
    const float sn =  cth;

    // normalized x for this thread's column in phase A: (x - 127.5)/127.5
    const float xxn = ((float)tid - 127.5f) * (1.0f / 127.5f);

    v8f acc[4];
    #pragma unroll
    for (int i = 0; i < 4; ++i) acc[i] = (v8f){0.f, 0.f, 0.f, 0.f, 0.f, 0.f, 0.f, 0.f};

    for (int y = y0; y < y1; ++y) {
        const int buf = y & 1;

        // ---------- Phase A: rotate row y of all 8 slices into LDS ----------
        {
            const float yyn = ((float)y - 127.5f) * (1.0f / 127.5f);
            const float rx = -xxn * sn + yyn * cs;
            const float ry =  xxn * cs + yyn * sn;
            const float xs  = (rx + 1.0f) * 127.5f;   // align_corners=True
            const float ysm = (ry + 1.0f) * 127.5f;
            const float x0f = floorf(xs);
            const float y0f = floorf(ysm);
            const float wx = xs - x0f;
            const float wy = ysm - y0f;
            const int xi0 = (int)x0f, yi0 = (int)y0f;
            const int xi1 = xi0 + 1,  yi1 = yi0 + 1;
            const float okx0 = (xi0 >= 0 && xi0 < NXE) ? 1.0f : 0.0f;
            const float okx1 = (xi1 >= 0 && xi1 < NXE) ? 1.0f : 0.0f;
            const float oky0 = (yi0 >= 0 && yi0 < NYE) ? 1.0f : 0.0f;
            const float oky1 = (yi1 >= 0 && yi1 < NYE) ? 1.0f : 0.0f;
            const int cx0 = min(max(xi0, 0), NXE - 1);
            const int cx1 = min(max(xi1, 0), NXE - 1);
            const int cy0 = min(max(yi0, 0), NYE - 1);
            const int cy1 = min(max(yi1, 0), NYE - 1);
            const float w00 = (1.0f - wy) * (1.0f - wx) * oky0 * okx0;
            const float w01 = (1.0f - wy) * wx          * oky0 * okx1;
            const float w10 = wy          * (1.0f - wx) * oky1 * okx0;
            const float w11 = wy          * wx          * oky1 * okx1;
            const int o00 = cy0 * NXE + cx0;
            const int o01 = cy0 * NXE + cx1;
            const int o10 = cy1 * NXE + cx0;
            const int o11 = cy1 * NXE + cx1;
            #pragma unroll
            for (int z = 0; z < NZE; ++z) {
                const float* img = vol + z * (NYE * NXE);
                P[buf][z][tid] = img[o00] * w00 + img[o01] * w01 +
                                 img[o10] * w10 + img[o11] * w11;
            }
        }
        __syncthreads();
        // (No trailing barrier: phase A of y+1 writes buffer (y+1)&1, whose last
        // readers -- phase B of y-1 -- all completed before this barrier.)

        // ---------- Phase B: projection as WMMA GEMM K-slices ----------
        const float ratio = (595.0f - ((float)y - 127.5f) * 1.953125f) * (1.0f / 1085.6f);

        // A-fragment: z-interp weights W[v,z]; zcoord = 3.5 + (v-3.5)*ratio (interior).
        // A 16x4 layout: VGPR0 = K0 (lanes 0-15) / K2 (lanes 16-31); VGPR1 = K1 / K3.
        const float vrow = (float)nidx;
        const float zc  = 3.5f + (vrow - 3.5f) * ratio;
        const float z0f = floorf(zc);
        const int   z0  = (int)z0f;
        const float wz  = zc - z0f;
        const float mrow = (nidx < NVE) ? 1.0f : 0.0f;   // rows 8..15 of A are zero
        v2f afrag[2];
        #pragma unroll
        for (int s = 0; s < 2; ++s) {
            const int zi = s * 4 + half * 2;             // z of component .x
            const float wA = (zi     == z0) ? (1.0f - wz) : ((zi     == z0 + 1) ? wz : 0.0f);
            const float wB = (zi + 1 == z0) ? (1.0f - wz) : ((zi + 1 == z0 + 1) ? wz : 0.0f);
            afrag[s].x = wA * mrow;
            afrag[s].y = wB * mrow;
        }

        // B-fragments + WMMA. xcoord = 127.5 + (u-255.5)*ratio*0.512 (interior).
        #pragma unroll
        for (int i = 0; i < 4; ++i) {
            const int u = (wave * 4 + i) * 16 + nidx;
            const float xc = 127.5f + ((float)u - 255.5f) * (ratio * 0.512f);
            const float xf = floorf(xc);
            const int x0p = (int)xf;
            const float wxp = xc - xf;
            const float wxm = 1.0f - wxp;
            #pragma unroll
            for (int s = 0; s < 2; ++s) {
                const int zb = s * 4 + half * 2;
                v2f bfrag;
                bfrag.x = P[buf][zb    ][x0p] * wxm + P[buf][zb    ][x0p + 1] * wxp;
                bfrag.y = P[buf][zb + 1][x0p] * wxm + P[buf][zb + 1][x0p + 1] * wxp;
                acc[i] = __builtin_amdgcn_wmma_f32_16x16x4_f32(
                    false, afrag[s], false, bfrag, (short)0, acc[i], false, false);
            }
        }
    }

    // ---------- Epilogue: store raw partial sums ----------
    // D layout: VGPR r, lanes 0-15 -> (v=r, u=u_base+lane); lanes 16-31 hold v=r+8 (pad).
    if (half == 0) {
        float* dst = partial + ((size_t)c * NAE + a) * (NVE * NUE);
        #pragma unroll
        for (int i = 0; i < 4; ++i) {
            const int u = (wave * 4 + i) * 16 + nidx;
            #pragma unroll
            for (int r = 0; r < NVE; ++r) {
                dst[r * NUE + u] = acc[i][r];
            }
        }
    }
}

// ---------------------------------------------------------------------------
// Reduce kernel: out[idx] = DIST[v,u] * sum_c partial[c*262144 + idx].
// Fixed-order sum over c -> deterministic. Safe in place when partial == out
// and nchunk == 1 (each thread touches only its own element).
// ---------------------------------------------------------------------------
__global__ __launch_bounds__(256) void fdk_reduce_kernel(const float* __restrict__ partial,
                                                         float* __restrict__ out,
                                                         int nchunk) {
    const int idx = blockIdx.x * 256 + threadIdx.x;      // over NA*NV*NU = 262144
    const int u = idx & (NUE - 1);
    const int v = (idx >> 9) & (NVE - 1);
    float s = 0.0f;
    for (int cc = 0; cc < nchunk; ++cc)
        s += partial[(size_t)cc * (NAE * NVE * NUE) + idx];
    const float du = (float)u - 255.5f;
    const float dv = (float)v - 3.5f;
    // DIST = sqrt(DSD^2 + u^2 + v^2) / DSD * DY
    const float dist = sqrtf(1178527.36f + du * du + dv * dv) * (1.953125f / 1085.6f);
    out[idx] = s * dist;
}

extern "C" void kernel_launch(void* const* d_in, const int* in_sizes, int n_in,
                              void* d_out, int out_size, void* d_ws, size_t ws_size,
                              hipStream_t stream) {
    const float* x = (const float*)d_in[0];
    float* out = (float*)d_out;
    (void)in_sizes; (void)n_in; (void)out_size;

    const size_t chunkBytes = (size_t)NAE * NVE * NUE * sizeof(float);  // 1 MB

    int nchunk = 1;
    float* base = out;   // fallback: raw sums straight into d_out, scaled in place
    if (d_ws != nullptr) {
        const int candidates[4] = {16, 8, 4, 2};
        for (int k = 0; k < 4; ++k) {
            if (ws_size >= (size_t)candidates[k] * chunkBytes) {
                nchunk = candidates[k];
                base = (float*)d_ws;
                break;
            }
        }
    }

    fdk_chunk_kernel<<<dim3(NAE, nchunk), 256, 0, stream>>>(x, base, NYE / nchunk);
    fdk_reduce_kernel<<<(NAE * NVE * NUE) / 256, 256, 0, stream>>>(base, out, nchunk);
}